// BaselineLoss_1168231104577
// MI455X (gfx1250) — compile-verified
//
#include <hip/hip_runtime.h>
#include <hip/hip_bf16.h>
#include <math.h>

typedef unsigned int u32;
typedef __attribute__((ext_vector_type(2))) float v2f;
typedef __attribute__((ext_vector_type(8))) float v8f;

#define BLOCK 256
#define ITEMS 4
#define TILE  (BLOCK * ITEMS)   // 1024
#define RADIX 256
#define NPASS 4

// ---------------- generic block inclusive scan (Hillis-Steele in LDS) -------
template <typename T>
__device__ __forceinline__ T blockInclusiveScan(T v, T* tmp) {
    int t = threadIdx.x;
    tmp[t] = v;
    __syncthreads();
    for (int off = 1; off < BLOCK; off <<= 1) {
        T x = (t >= off) ? tmp[t - off] : (T)0;
        __syncthreads();
        tmp[t] += x;
        __syncthreads();
    }
    T r = tmp[t];
    __syncthreads();
    return r;
}

// ---------------- 0) build descending-stable sort keys -----------------------
// ascending-orderable map: u = (b>>31) ? ~b : (b | 0x80000000); descending key = ~u
__global__ void k_build_keys(const float* __restrict__ truth,
                             u32* __restrict__ keys, u32* __restrict__ idx, int n) {
    int j = blockIdx.x * BLOCK + threadIdx.x;
    if (j < n) {
        u32 b = __float_as_uint(truth[2 * (size_t)j + 1]);   // time
        u32 u = (b & 0x80000000u) ? ~b : (b | 0x80000000u);
        keys[j] = ~u;
        idx[j]  = (u32)j;
    }
}

// ---------------- 1) per-block digit histogram (async LDS staging) ----------
__global__ void k_hist(const u32* __restrict__ keys, u32* __restrict__ histG,
                       int shift, int nb) {
    __shared__ u32 h[RADIX];
    __shared__ u32 stage[TILE];
    int t = threadIdx.x, b = blockIdx.x;
    h[t] = 0;
    const u32* src = keys + (size_t)b * TILE;
#if defined(__gfx1250__) && __has_builtin(__builtin_amdgcn_global_load_async_to_lds_b32)
    {
        u32* gsrc = const_cast<u32*>(src);
        int w = t >> 5, l = t & 31;
        #pragma unroll
        for (int i = 0; i < ITEMS; ++i) {
            int j = w * (32 * ITEMS) + i * 32 + l;
            __builtin_amdgcn_global_load_async_to_lds_b32(
                (__attribute__((address_space(1))) int*)(gsrc + j),
                (__attribute__((address_space(3))) int*)&stage[j],
                0, 0);
        }
#if __has_builtin(__builtin_amdgcn_s_wait_asynccnt)
        __builtin_amdgcn_s_wait_asynccnt(0);
#else
        asm volatile("s_wait_asynccnt 0" ::: "memory");
#endif
    }
#else
    for (int i = 0; i < ITEMS; ++i) { int j = i * BLOCK + t; stage[j] = src[j]; }
#endif
    __syncthreads();
    for (int i = 0; i < ITEMS; ++i) {
        u32 k = stage[i * BLOCK + t];
        atomicAdd(&h[(k >> shift) & (RADIX - 1)], 1u);
    }
    __syncthreads();
    // digit-major layout so the per-digit scan reads contiguously
    histG[(size_t)t * nb + b] = h[t];
}

// ---------------- 2) per-digit exclusive scan over blocks --------------------
__global__ void k_scan_hist(u32* __restrict__ histG, u32* __restrict__ digitTotal,
                            int nb) {
    __shared__ u32 tmp[BLOCK];
    __shared__ u32 carry;
    int d = blockIdx.x, t = threadIdx.x;
    if (t == 0) carry = 0;
    __syncthreads();
    u32* h = histG + (size_t)d * nb;
    for (int base = 0; base < nb; base += BLOCK) {
        u32 c0 = carry;
        u32 v = h[base + t];
        u32 inc = blockInclusiveScan(v, tmp);
        __syncthreads();
        if (t == BLOCK - 1) carry = c0 + inc;
        h[base + t] = c0 + inc - v;          // exclusive
        __syncthreads();
    }
    if (t == 0) digitTotal[d] = carry;
}

// ---------------- 3) exclusive scan of the 256 digit totals ------------------
__global__ void k_scan_digits(const u32* __restrict__ digitTotal,
                              u32* __restrict__ digitBase) {
    __shared__ u32 tmp[BLOCK];
    int t = threadIdx.x;
    u32 v = digitTotal[t];
    u32 inc = blockInclusiveScan(v, tmp);
    digitBase[t] = inc - v;
}

// ---------------- 4) stable ballot-ranked scatter ----------------------------
// element order inside a block: j = wave*128 + slot*32 + lane; ranks computed
// in (wave, slot, lane) order via wave-serialized shared counters => stable.
__global__ void k_scatter(const u32* __restrict__ keyIn, const u32* __restrict__ idxIn,
                          u32* __restrict__ keyOut, u32* __restrict__ idxOut,
                          const u32* __restrict__ histG, const u32* __restrict__ digitBase,
                          int shift, int nb) {
    __shared__ u32 cnt[RADIX];
    __shared__ u32 baseD[RADIX];
    int t = threadIdx.x, b = blockIdx.x;
    cnt[t] = 0;
    baseD[t] = digitBase[t] + histG[(size_t)t * nb + b];
    __syncthreads();
    int wv = t >> 5, lane = t & 31;
    size_t start = (size_t)b * TILE;
    u32 key[ITEMS], idxv[ITEMS], rank[ITEMS];
    #pragma unroll
    for (int i = 0; i < ITEMS; ++i) {
        int j = wv * (32 * ITEMS) + i * 32 + lane;
        key[i]  = keyIn[start + j];
        idxv[i] = idxIn[start + j];
    }
    u32 lt = (1u << lane) - 1u;
    for (int w = 0; w < BLOCK / 32; ++w) {
        if (wv == w) {
            #pragma unroll
            for (int i = 0; i < ITEMS; ++i) {
                u32 d = (key[i] >> shift) & (RADIX - 1);
                u32 match = 0xFFFFFFFFu;
                #pragma unroll
                for (int bit = 0; bit < 8; ++bit) {
                    bool p = (d >> bit) & 1u;
                    u32 vote = (u32)__ballot(p);          // wave32 ballot
                    match &= p ? vote : ~vote;
                }
                u32 lr = __popc(match & lt);
                int leader = __ffs(match) - 1;
                u32 base_ = 0;
                if ((int)lane == leader) base_ = atomicAdd(&cnt[d], (u32)__popc(match));
                base_ = __shfl(base_, leader);
                rank[i] = base_ + lr;
            }
        }
        __syncthreads();   // enforce wave order -> stability
    }
    #pragma unroll
    for (int i = 0; i < ITEMS; ++i) {
        u32 d = (key[i] >> shift) & (RADIX - 1);
        u32 pos = baseD[d] + rank[i];
        keyOut[pos] = key[i];
        idxOut[pos] = idxv[i];
    }
}

// ---------------- 5) global max of scores (for logsumexp shift) --------------
__global__ void k_max1(const float* __restrict__ scores, float* __restrict__ part) {
    __shared__ float tmp[BLOCK];
    int t = threadIdx.x, b = blockIdx.x;
    size_t s0 = (size_t)b * TILE;
    float m = -INFINITY;
    for (int i = 0; i < ITEMS; ++i) m = fmaxf(m, scores[s0 + i * BLOCK + t]);
    tmp[t] = m; __syncthreads();
    for (int off = 128; off > 0; off >>= 1) { if (t < off) tmp[t] = fmaxf(tmp[t], tmp[t + off]); __syncthreads(); }
    if (t == 0) part[b] = tmp[0];
}
__global__ void k_max2(const float* __restrict__ part, int n, float* __restrict__ Mout) {
    __shared__ float tmp[BLOCK];
    int t = threadIdx.x;
    float m = -INFINITY;
    for (int j = t; j < n; j += BLOCK) m = fmaxf(m, part[j]);
    tmp[t] = m; __syncthreads();
    for (int off = 128; off > 0; off >>= 1) { if (t < off) tmp[t] = fmaxf(tmp[t], tmp[t + off]); __syncthreads(); }
    if (t == 0) Mout[0] = tmp[0];
}

// ---------------- 6) block partial sums of exp(s-M) in sorted order ----------
__global__ void k_scan1(const u32* __restrict__ idxSorted, const float* __restrict__ scores,
                        const float* __restrict__ Mp, float* __restrict__ partials) {
    __shared__ float tmp[BLOCK];
    int t = threadIdx.x, b = blockIdx.x;
    float M = Mp[0];
    const uint4* p = (const uint4*)(idxSorted + (size_t)b * TILE);
    uint4 id = p[t];
    float s = expf(scores[id.x] - M) + expf(scores[id.y] - M) +
              expf(scores[id.z] - M) + expf(scores[id.w] - M);
    tmp[t] = s; __syncthreads();
    for (int off = 128; off > 0; off >>= 1) { if (t < off) tmp[t] += tmp[t + off]; __syncthreads(); }
    if (t == 0) partials[b] = tmp[0];
}

// ---------------- 7) exclusive scan of block partials (single block) ---------
__global__ void k_scan_partials(float* __restrict__ partials, int n) {
    __shared__ float tmp[BLOCK];
    __shared__ float carry;
    int t = threadIdx.x;
    if (t == 0) carry = 0.f;
    __syncthreads();
    for (int base = 0; base < n; base += BLOCK) {
        float c0 = carry;
        float v = partials[base + t];
        float inc = blockInclusiveScan(v, tmp);
        __syncthreads();
        if (t == BLOCK - 1) carry = c0 + inc;
        partials[base + t] = c0 + inc - v;   // exclusive
        __syncthreads();
    }
}

// ---------------- 8) WMMA-accelerated block prefix + loss terms --------------
// 256 per-thread sums -> 16x16 matrix X (col = segment of 16 threads).
// P = L(lower-tri ones) x X via 4 chained V_WMMA_F32_16X16X4_F32 gives the
// intra-segment inclusive prefixes; row 15 gives segment sums.
__global__ void k_scan3(const u32* __restrict__ idxSorted, const float* __restrict__ scores,
                        const float* __restrict__ truth, const float* __restrict__ Mp,
                        const float* __restrict__ partExcl, float* __restrict__ lossPart) {
    __shared__ float sums[BLOCK];
    __shared__ float pmat[256];
    int t = threadIdx.x, b = blockIdx.x;
    float M = Mp[0];
    const uint4* p = (const uint4*)(idxSorted + (size_t)b * TILE);
    uint4 id = p[t];
    u32 ids[4] = { id.x, id.y, id.z, id.w };
    float sv[4], ev[4], xv[4];
    #pragma unroll
    for (int i = 0; i < 4; ++i) {
        sv[i] = scores[ids[i]];
        ev[i] = truth[2 * (size_t)ids[i]];   // event flag
        xv[i] = expf(sv[i] - M);
    }
    float tSum = xv[0] + xv[1] + xv[2] + xv[3];
    sums[t] = tSum;
    __syncthreads();
    if ((t >> 5) == 0) {                       // wave 0 only (uniform branch, EXEC all-1)
        int lane = t;
        int half = lane >> 4, lm = lane & 15;  // A: m=lm ; B/D: n=lm
        v8f acc = {};
        #pragma unroll
        for (int c = 0; c < 4; ++c) {
            int kg = 4 * c + 2 * half;         // global K for this lane's 2 slots
            v2f a, bb;
            a.x = (kg     <= lm) ? 1.f : 0.f;  // L[m][kg]
            a.y = (kg + 1 <= lm) ? 1.f : 0.f;  // L[m][kg+1]
            bb.x = sums[16 * lm + kg];         // X[kg][n]   = v[16n+kg]
            bb.y = sums[16 * lm + kg + 1];     // X[kg+1][n]
            acc = __builtin_amdgcn_wmma_f32_16x16x4_f32(
                false, a, false, bb, (short)0, acc, false, false);
        }
        #pragma unroll
        for (int r = 0; r < 8; ++r)            // D: vgpr r = rows r, r+8
            pmat[(r + 8 * half) * 16 + lm] = acc[r];
    }
    __syncthreads();
    int cseg = t >> 4, k = t & 15;
    float incl = pmat[k * 16 + cseg];          // inclusive within my segment
    float segOff = 0.f;
    for (int cc = 0; cc < cseg; ++cc) segOff += pmat[240 + cc];  // row 15 = segment sums
    float run = partExcl[b] + segOff + incl - tSum;              // exclusive for thread t
    float acc2 = 0.f;
    #pragma unroll
    for (int i = 0; i < 4; ++i) {
        run += xv[i];
        acc2 += (logf(run) + M - sv[i]) * ev[i];
    }
    __syncthreads();
    sums[t] = acc2; __syncthreads();
    for (int off = 128; off > 0; off >>= 1) { if (t < off) sums[t] += sums[t + off]; __syncthreads(); }
    if (t == 0) lossPart[b] = sums[0];
}

// ---------------- 9) deterministic final reduction ---------------------------
__global__ void k_reduce(const float* __restrict__ part, int n, float invN,
                         float* __restrict__ out) {
    __shared__ float tmp[BLOCK];
    int t = threadIdx.x;
    float s = 0.f;
    for (int j = t; j < n; j += BLOCK) s += part[j];
    tmp[t] = s; __syncthreads();
    for (int off = 128; off > 0; off >>= 1) { if (t < off) tmp[t] += tmp[t + off]; __syncthreads(); }
    if (t == 0) out[0] = tmp[0] * invN;
}

// ---------------- host-side launcher ----------------------------------------
extern "C" void kernel_launch(void* const* d_in, const int* in_sizes, int n_in,
                              void* d_out, int out_size, void* d_ws, size_t ws_size,
                              hipStream_t stream) {
    const float* scores = (const float*)d_in[0];   // (N,1) f32
    const float* truth  = (const float*)d_in[1];   // (N,2) f32 [event,time]
    int n  = in_sizes[0];                          // 8,388,608 (multiple of TILE)
    int nb = n / TILE;                             // 8192

    char* w = (char*)d_ws;
    auto alloc = [&](size_t bytes) -> char* {
        char* p = w; w += (bytes + 255) & ~(size_t)255; return p;
    };
    u32*   keysA      = (u32*)  alloc((size_t)n * 4);
    u32*   idxA       = (u32*)  alloc((size_t)n * 4);
    u32*   keysB      = (u32*)  alloc((size_t)n * 4);
    u32*   idxB       = (u32*)  alloc((size_t)n * 4);
    u32*   histG      = (u32*)  alloc((size_t)RADIX * nb * 4);
    u32*   digitTotal = (u32*)  alloc(RADIX * 4);
    u32*   digitBase  = (u32*)  alloc(RADIX * 4);
    float* maxPart    = (float*)alloc((size_t)nb * 4);
    float* Mval       = (float*)alloc(256);
    float* partials   = (float*)alloc((size_t)nb * 4);
    float* lossPart   = (float*)alloc((size_t)nb * 4);
    (void)ws_size; (void)n_in; (void)out_size;

    k_build_keys<<<n / BLOCK, BLOCK, 0, stream>>>(truth, keysA, idxA, n);

    u32 *ki = keysA, *ii = idxA, *ko = keysB, *io = idxB;
    for (int pass = 0; pass < NPASS; ++pass) {
        int shift = 8 * pass;
        k_hist       <<<nb,    BLOCK, 0, stream>>>(ki, histG, shift, nb);
        k_scan_hist  <<<RADIX, BLOCK, 0, stream>>>(histG, digitTotal, nb);
        k_scan_digits<<<1,     BLOCK, 0, stream>>>(digitTotal, digitBase);
        k_scatter    <<<nb,    BLOCK, 0, stream>>>(ki, ii, ko, io, histG, digitBase, shift, nb);
        u32* tk = ki; ki = ko; ko = tk;
        u32* ti = ii; ii = io; io = ti;
    }
    // after 4 passes, sorted keys/indices are back in keysA/idxA (== ki/ii)

    k_max1<<<nb, BLOCK, 0, stream>>>(scores, maxPart);
    k_max2<<<1,  BLOCK, 0, stream>>>(maxPart, nb, Mval);

    k_scan1        <<<nb, BLOCK, 0, stream>>>(ii, scores, Mval, partials);
    k_scan_partials<<<1,  BLOCK, 0, stream>>>(partials, nb);
    k_scan3        <<<nb, BLOCK, 0, stream>>>(ii, scores, truth, Mval, partials, lossPart);
    k_reduce       <<<1,  BLOCK, 0, stream>>>(lossPart, nb, 1.0f / (float)n, (float*)d_out);
}